// DisplacementLoss_46557445488917
// MI455X (gfx1250) — compile-verified
//
#include <hip/hip_runtime.h>

typedef __attribute__((ext_vector_type(2))) float v2f;
typedef __attribute__((ext_vector_type(8))) float v8f;

#define B_DIM 4
#define N_PTS 8192
#define KNN 16
#define TILE 16
#define NTILES (N_PTS / TILE)                 // 512 candidate tiles
#define WAVES_PER_BLOCK 8
#define THREADS (WAVES_PER_BLOCK * 32)        // 256 threads, 8 wave32
#define TOTAL_TILES (B_DIM * NTILES)          // 2048 query tiles
#define NBLOCKS (TOTAL_TILES / WAVES_PER_BLOCK) // 256 blocks

// Fused kNN displacement loss.
// Per wave: 16 query points. A = streaming candidate tile (16x4: x,y,z,|p|^2),
// B = fixed queries (4x16: -2x,-2y,-2z,1), C = |q|^2 per lane (loop-invariant).
// D[m][n] = |p_m|^2 + |q_n|^2 - 2 p.q  ==> lane n%16 gets 8 distances for ITS
// query straight in registers: no LDS transpose in the hot loop. Candidate
// loads are software-pipelined one tile ahead to hide L2 latency.
__global__ __launch_bounds__(THREADS) void knn_disp_kernel(
    const float* __restrict__ adv, const float* __restrict__ ori,
    float* __restrict__ partial)
{
    // Per-wave merge scratch only: 512 dists + 512 idx = 4KB/wave.
    __shared__ float stage[WAVES_PER_BLOCK * 1024];

    const int tid  = threadIdx.x;
    const int wave = tid >> 5;
    const int lane = tid & 31;
    const int half = lane >> 4;    // 0: lanes 0-15, 1: lanes 16-31
    const int lm   = lane & 15;

    const int tileId = blockIdx.x * WAVES_PER_BLOCK + wave;
    const int b      = tileId / NTILES;
    const int row0   = (tileId % NTILES) * TILE;

    const float* orib = ori + (size_t)b * N_PTS * 3;
    const float* advb = adv + (size_t)b * N_PTS * 3;
    float* mystage = stage + wave * 1024;

    // ---- Loop-invariant B fragment from the 16 query points ----
    // B 4x16 layout: lanes 0-15 hold K0(v0),K1(v1) at N=lm; lanes 16-31 hold K2,K3.
    const int qi = row0 + lm;
    const float qx = orib[qi * 3 + 0];
    const float qy = orib[qi * 3 + 1];
    const float qz = orib[qi * 3 + 2];
    const float qsq = qx * qx + qy * qy + qz * qz;
    v2f bfrag;
    bfrag.x = half ? (-2.0f * qz) : (-2.0f * qx);
    bfrag.y = half ? 1.0f         : (-2.0f * qy);

    // Loop-invariant C fragment: |q_n|^2 (C layout: lane -> column N = query).
    v8f cfrag;
#pragma unroll
    for (int r = 0; r < 8; ++r) cfrag[r] = qsq;

    // Register-resident sorted (ascending) top-16 per lane (half of one query's
    // candidate stream: lane l covers candidate rows 8*half..8*half+7 per tile).
    float hd[KNN];
    int   hi[KNN];
#pragma unroll
    for (int s = 0; s < KNN; ++s) { hd[s] = 3.4e38f; hi[s] = 0; }

    // Software pipeline: preload candidate point of tile 0.
    float px = orib[lm * 3 + 0];
    float py = orib[lm * 3 + 1];
    float pz = orib[lm * 3 + 2];

#pragma unroll 2
    for (int ct = 0; ct < NTILES; ++ct) {
        // ---- Prefetch next tile's candidate point (wraps to tile 0 at end) ----
        const int nj = (((ct + 1) & (NTILES - 1)) * TILE + lm) * 3;
        const float nx = orib[nj + 0];
        const float ny = orib[nj + 1];
        const float nz = orib[nj + 2];

        // ---- A fragment: current candidate tile, K = (x, y, z, |p|^2) ----
        // A 16x4 layout: lanes 0-15 -> M=lm holds K0(v0),K1(v1); lanes 16-31 K2,K3.
        const float psq = px * px + py * py + pz * pz;
        v2f afrag;
        afrag.x = half ? pz  : px;
        afrag.y = half ? psq : py;

        v8f dfrag = __builtin_amdgcn_wmma_f32_16x16x4_f32(
            false, afrag, false, bfrag, (short)0, cfrag, false, false);

        // D layout: VGPR r, lane l -> D[M = r + 8*(l>=16)][N = l%16].
        // So dfrag[r] is dist(candidate ct*16 + 8*half + r, query row0+lm).
#pragma unroll
        for (int r = 0; r < 8; ++r) {
            const float dv = dfrag[r];
            const int   dj = ct * TILE + 8 * half + r;
            if (dv < hd[KNN - 1]) {
                // Branchless unrolled sorted insert (all static indices -> VGPRs).
#pragma unroll
                for (int s = KNN - 1; s >= 1; --s) {
                    const bool shift = dv < hd[s - 1];
                    const bool here  = !shift && (dv < hd[s]);
                    const float nhd = shift ? hd[s - 1] : (here ? dv : hd[s]);
                    const int   nhi = shift ? hi[s - 1] : (here ? dj : hi[s]);
                    hd[s] = nhd; hi[s] = nhi;
                }
                if (dv < hd[0]) { hd[0] = dv; hi[0] = dj; }
            }
        }

        px = nx; py = ny; pz = nz;
    }

    // ---- Merge the two half-lists per query via LDS ----
    // Lane l (<16) has candidates {m%16 in 0..7}, lane l+16 has {8..15}, same query.
#pragma unroll
    for (int s = 0; s < KNN; ++s) {
        mystage[lane * 16 + s] = hd[s];
        ((int*)mystage)[512 + lane * 16 + s] = hi[s];
    }
    __builtin_amdgcn_wave_barrier();

    float acc = 0.0f;
    if (lane < 16) {
        const float* Ad = mystage + lane * 16;
        const float* Bd = mystage + (lane + 16) * 16;
        const int*   Ai = ((const int*)mystage) + 512 + lane * 16;
        const int*   Bi = ((const int*)mystage) + 512 + (lane + 16) * 16;
        int knn_idx[KNN];
        int ia = 0, ib = 0;
#pragma unroll
        for (int t = 0; t < KNN; ++t) {
            const float da = (ia < KNN) ? Ad[ia] : 3.5e38f;
            const float db = (ib < KNN) ? Bd[ib] : 3.5e38f;
            if (da <= db) { knn_idx[t] = Ai[ia]; ++ia; }
            else          { knn_idx[t] = Bi[ib]; ++ib; }
        }

        // ---- Displacement loss for query (row0 + lane) ----
        const int i = row0 + lane;
        const float dix = advb[i * 3 + 0] - orib[i * 3 + 0];
        const float diy = advb[i * 3 + 1] - orib[i * 3 + 1];
        const float diz = advb[i * 3 + 2] - orib[i * 3 + 2];
#pragma unroll
        for (int t = 0; t < KNN; ++t) {
            const int jn = knn_idx[t];
            const float djx = advb[jn * 3 + 0] - orib[jn * 3 + 0];
            const float djy = advb[jn * 3 + 1] - orib[jn * 3 + 1];
            const float djz = advb[jn * 3 + 2] - orib[jn * 3 + 2];
            const float ex = dix - djx, ey = diy - djy, ez = diz - djz;
            acc += ex * ex + ey * ey + ez * ez;
        }
    }

    // Wave reduction (lanes >=16 contribute 0), deterministic per tile.
#pragma unroll
    for (int off = 16; off >= 1; off >>= 1)
        acc += __shfl_xor(acc, off, 32);
    if (lane == 0) partial[tileId] = acc;
}

__global__ __launch_bounds__(256) void knn_disp_reduce(
    const float* __restrict__ partial, float* __restrict__ out)
{
    __shared__ float sbuf[256];
    const int t = threadIdx.x;
    float v = 0.0f;
    for (int i = t; i < TOTAL_TILES; i += 256) v += partial[i];
    sbuf[t] = v;
    __syncthreads();
    for (int off = 128; off >= 1; off >>= 1) {
        if (t < off) sbuf[t] += sbuf[t + off];
        __syncthreads();
    }
    if (t == 0)
        out[0] = sbuf[0] * (1.0f / (float)(B_DIM * (size_t)N_PTS * KNN));
}

extern "C" void kernel_launch(void* const* d_in, const int* in_sizes, int n_in,
                              void* d_out, int out_size, void* d_ws, size_t ws_size,
                              hipStream_t stream) {
    const float* adv = (const float*)d_in[0];
    const float* ori = (const float*)d_in[1];
    // d_in[2] is k (==16 on device); KNN is compiled in to keep top-k in registers.
    float* partial = (float*)d_ws;            // 2048 floats of scratch
    float* out = (float*)d_out;

    knn_disp_kernel<<<NBLOCKS, THREADS, 0, stream>>>(adv, ori, partial);
    knn_disp_reduce<<<1, 256, 0, stream>>>(partial, out);
}